// WindowOrthogonalMamba_59064390255106
// MI455X (gfx1250) — compile-verified
//
#include <hip/hip_runtime.h>
#include <hip/hip_bf16.h>
#include <stdint.h>

#define BATCH 16
#define WS    64
#define L     (WS*WS)     // 4096
#define CH    256
#define NDIR  4
#define LN_EPS 1e-5f

// ws layout: r[4][16][64][256], then sInit[4][16][64][256]  (4 MB each)
#define DSTRIDE (BATCH*WS*CH)   // per-direction stride in floats

typedef unsigned int  u32x4 __attribute__((ext_vector_type(4)));
typedef int           i32x8 __attribute__((ext_vector_type(8)));
typedef int           i32x4 __attribute__((ext_vector_type(4)));

#if defined(__has_builtin)
#  if __has_builtin(__builtin_amdgcn_tensor_load_to_lds)
#    define HAVE_TDM 1
#  endif
#endif

// ---------------------------------------------------------------------------
// Async global->LDS tile copy (64 rows x 256 ch fp32 = 64KB), CDNA5 async op.
// Each of 256 threads issues 16 x b128 transfers. LDS offset = low 32 bits of
// the flat shared-aperture pointer (ISA 10.2).
// ---------------------------------------------------------------------------
__device__ __forceinline__ void async_tile_load(const float* __restrict__ g,
                                                float* __restrict__ l) {
    const int t = threadIdx.x;
#pragma unroll
    for (int i = 0; i < 16; ++i) {
        const int idx = i * 1024 + t * 4;             // floats; 16B per issue
        unsigned           lo = (unsigned)(uintptr_t)(l + idx);
        unsigned long long go = (unsigned long long)(uintptr_t)(g + idx);
        asm volatile("global_load_async_to_lds_b128 %0, %1, off"
                     :: "v"(lo), "v"(go) : "memory");
    }
}

__device__ __forceinline__ float wred(float v) {   // wave32 sum
    v += __shfl_xor(v, 1, 32);
    v += __shfl_xor(v, 2, 32);
    v += __shfl_xor(v, 4, 32);
    v += __shfl_xor(v, 8, 32);
    v += __shfl_xor(v, 16, 32);
    return v;
}

// ---------------------------------------------------------------------------
// K1: per (b, row k): 64-step from-zero scan for all 4 directions -> residual r
// ---------------------------------------------------------------------------
__global__ __launch_bounds__(256)
void k_chunk(const float* __restrict__ w, const float* __restrict__ A,
             const float* __restrict__ Bp, float* __restrict__ r) {
    const int c = threadIdx.x;
    const int k = blockIdx.x;          // window row / chunk
    const int b = blockIdx.y;
    const float a  = A[c];
    const float bp = Bp[c];
    const float* rowA = w + ((size_t)b * L + (size_t)WS * k)            * CH + c;
    const float* rowB = w + ((size_t)b * L + (size_t)WS * (WS - 1 - k)) * CH + c;

    float s1 = 0.f, s2 = 0.f, s3 = 0.f, s4 = 0.f;
#pragma unroll 4
    for (int col = 0; col < WS; ++col) {
        const float x1 = rowA[(size_t)col * CH];
        const float x3 = rowA[(size_t)(WS - 1 - col) * CH];
        const float x4 = rowB[(size_t)col * CH];
        const float x2 = rowB[(size_t)(WS - 1 - col) * CH];
        s1 = fmaf(a, s1, bp * x1);
        s2 = fmaf(a, s2, bp * x2);
        s3 = fmaf(a, s3, bp * x3);
        s4 = fmaf(a, s4, bp * x4);
    }
    const size_t base = ((size_t)b * WS + k) * CH + c;
    r[0 * (size_t)DSTRIDE + base] = s1;
    r[1 * (size_t)DSTRIDE + base] = s2;
    r[2 * (size_t)DSTRIDE + base] = s3;
    r[3 * (size_t)DSTRIDE + base] = s4;
}

// ---------------------------------------------------------------------------
// K2: per (dir,b,channel): prefix-combine chunk residuals -> per-chunk init
// ---------------------------------------------------------------------------
__global__ __launch_bounds__(256)
void k_prefix(const float* __restrict__ A, const float* __restrict__ r,
              float* __restrict__ sInit) {
    const int c = threadIdx.x;
    const int d = blockIdx.x >> 4;
    const int b = blockIdx.x & 15;
    float a64 = A[c];
#pragma unroll
    for (int i = 0; i < 6; ++i) a64 *= a64;   // A^64
    const size_t base = ((size_t)d * BATCH + b) * (size_t)(WS * CH) + c;
    float s = 0.f;
    for (int k = 0; k < WS; ++k) {
        sInit[base + (size_t)k * CH] = s;
        s = fmaf(a64, s, r[base + (size_t)k * CH]);
    }
}

// ---------------------------------------------------------------------------
// K3: per (b, row k): TDM/async-load both tiles into LDS, rescan with true
// initial states, fused LayerNorm (wave32 shuffles + LDS), average, store.
// ---------------------------------------------------------------------------
__global__ __launch_bounds__(256)
void k_final(const float* __restrict__ w, const float* __restrict__ A,
             const float* __restrict__ Bp, const float* __restrict__ Cp,
             const float* __restrict__ Dp, const float* __restrict__ lnw,
             const float* __restrict__ lnb, const float* __restrict__ sInit,
             float* __restrict__ out) {
    __shared__ float tileA[WS * CH];   // row k        (dirs 1 fwd, 3 rev)
    __shared__ float tileB[WS * CH];   // row 63-k     (dirs 4 fwd, 2 rev)
    __shared__ float red[8][8];

    const int c = threadIdx.x;
    const int k = blockIdx.x;
    const int b = blockIdx.y;
    const float* gA = w + ((size_t)b * L + (size_t)WS * k)            * CH;
    const float* gB = w + ((size_t)b * L + (size_t)WS * (WS - 1 - k)) * CH;

#if HAVE_TDM
    // Tile A via Tensor Data Mover: 1-D transfer, 16384 x fp32 (64 KB).
    if (threadIdx.x < 32) {            // wave 0 only issues the DMA
        const unsigned long long ga = (unsigned long long)(uintptr_t)gA;
        const unsigned           la = (unsigned)(uintptr_t)&tileA[0];
        u32x4 g0;                       // D# group 0 (ISA 8.3)
        g0[0] = 1u;                                          // count=1
        g0[1] = la;                                          // lds_addr
        g0[2] = (unsigned)(ga & 0xFFFFFFFFu);                // global_addr lo
        g0[3] = ((unsigned)(ga >> 32) & 0x01FFFFFFu) | 0x80000000u; // hi | type=2
        i32x8 g1;                       // D# group 1 (ISA 8.4)
        g1[0] = 0x00020000;             // data_size=4B, mask=0
        g1[1] = (int)(16384u << 16);    // tensor_dim0 lo16 = 16384
        g1[2] = 0x00010000;             // tensor_dim0 hi=0 | tensor_dim1=1
        g1[3] = (int)(16384u << 16);    // tile_dim0 = 16384
        g1[4] = 1;                      // tile_dim1 = 1, tile_dim2 = 0
        g1[5] = 16384;                  // tensor_dim0_stride lo
        g1[6] = (int)(16384u << 16);    // stride hi=0 | tensor_dim1_stride lo16
        g1[7] = 0;
        i32x4 g2 = {0, 0, 0, 0};
        i32x4 g3 = {0, 0, 0, 0};
        i32x8 g4 = {0, 0, 0, 0, 0, 0, 0, 0};   // extra group (6-arg toolchain)
        __builtin_amdgcn_tensor_load_to_lds(g0, g1, g2, g3, g4, 0);
        __builtin_amdgcn_s_wait_tensorcnt(0);
    }
#else
    async_tile_load(gA, tileA);
#endif
    // Tile B via async global->LDS b128 (all 256 threads).
    async_tile_load(gB, tileB);
    asm volatile("s_wait_asynccnt 0" ::: "memory");
    __syncthreads();

    const float a  = A[c],  bp = Bp[c], cp = Cp[c], dp = Dp[c];
    const float lw = lnw[c], lb = lnb[c];
    const size_t ib = ((size_t)b * WS + k) * CH + c;
    float s1 = sInit[0 * (size_t)DSTRIDE + ib];
    float s2 = sInit[1 * (size_t)DSTRIDE + ib];
    float s3 = sInit[2 * (size_t)DSTRIDE + ib];
    float s4 = sInit[3 * (size_t)DSTRIDE + ib];

    float* outRow = out + ((size_t)b * L + (size_t)WS * k) * CH + c;
    const int lane = threadIdx.x & 31;
    const int wv   = threadIdx.x >> 5;
    const float inv = 1.0f / (float)CH;

    for (int col = 0; col < WS; ++col) {
        const float x1 = tileA[col * CH + c];
        const float x3 = tileA[(WS - 1 - col) * CH + c];
        const float x4 = tileB[col * CH + c];
        const float x2 = tileB[(WS - 1 - col) * CH + c];
        s1 = fmaf(a, s1, bp * x1);  const float y1 = fmaf(cp, s1, dp * x1);
        s2 = fmaf(a, s2, bp * x2);  const float y2 = fmaf(cp, s2, dp * x2);
        s3 = fmaf(a, s3, bp * x3);  const float y3 = fmaf(cp, s3, dp * x3);
        s4 = fmaf(a, s4, bp * x4);  const float y4 = fmaf(cp, s4, dp * x4);

        const float q0 = wred(y1), q1 = wred(y1 * y1);
        const float q2 = wred(y2), q3 = wred(y2 * y2);
        const float q4 = wred(y3), q5 = wred(y3 * y3);
        const float q6 = wred(y4), q7 = wred(y4 * y4);
        if (lane == 0) {
            red[wv][0] = q0; red[wv][1] = q1; red[wv][2] = q2; red[wv][3] = q3;
            red[wv][4] = q4; red[wv][5] = q5; red[wv][6] = q6; red[wv][7] = q7;
        }
        __syncthreads();
        float S1 = 0.f, Q1 = 0.f, S2 = 0.f, Q2 = 0.f;
        float S3 = 0.f, Q3 = 0.f, S4 = 0.f, Q4 = 0.f;
#pragma unroll
        for (int j = 0; j < 8; ++j) {
            S1 += red[j][0]; Q1 += red[j][1]; S2 += red[j][2]; Q2 += red[j][3];
            S3 += red[j][4]; Q3 += red[j][5]; S4 += red[j][6]; Q4 += red[j][7];
        }
        const float m1 = S1 * inv, m2 = S2 * inv, m3 = S3 * inv, m4 = S4 * inv;
        const float i1 = rsqrtf(fmaf(-m1, m1, Q1 * inv) + LN_EPS);
        const float i2 = rsqrtf(fmaf(-m2, m2, Q2 * inv) + LN_EPS);
        const float i3 = rsqrtf(fmaf(-m3, m3, Q3 * inv) + LN_EPS);
        const float i4 = rsqrtf(fmaf(-m4, m4, Q4 * inv) + LN_EPS);
        const float acc = (y1 - m1) * i1 + (y2 - m2) * i2 +
                          (y3 - m3) * i3 + (y4 - m4) * i4;
        outRow[(size_t)col * CH] = fmaf(0.25f * lw, acc, lb);
        __syncthreads();   // protect red[] for next iteration
    }
}

extern "C" void kernel_launch(void* const* d_in, const int* in_sizes, int n_in,
                              void* d_out, int out_size, void* d_ws, size_t ws_size,
                              hipStream_t stream) {
    const float* w   = (const float*)d_in[0];
    const float* A   = (const float*)d_in[1];
    const float* Bp  = (const float*)d_in[2];
    const float* Cp  = (const float*)d_in[3];
    const float* Dp  = (const float*)d_in[4];
    const float* lnw = (const float*)d_in[5];
    const float* lnb = (const float*)d_in[6];
    float* out = (float*)d_out;

    float* r     = (float*)d_ws;                          // 4 MB
    float* sInit = r + (size_t)NDIR * DSTRIDE;            // 4 MB

    dim3 grid2d(WS, BATCH);
    k_chunk <<<grid2d, 256, 0, stream>>>(w, A, Bp, r);
    k_prefix<<<NDIR * BATCH, 256, 0, stream>>>(A, r, sInit);
    k_final <<<grid2d, 256, 0, stream>>>(w, A, Bp, Cp, Dp, lnw, lnb, sInit, out);
}